// EvolveGCNH_13907104104626
// MI455X (gfx1250) — compile-verified
//
#include <hip/hip_runtime.h>

#define IN_FEATS 166
#define NH 76
#define TT 3
#define NN 100000
#define EE 1600000
#define CLSH 510
#define RRELU_SLOPE 0.22916666666666666f

typedef float v2f __attribute__((ext_vector_type(2)));
typedef float v8f __attribute__((ext_vector_type(8)));

__device__ __forceinline__ void fatomic_add(float* p, float v) {
  __hip_atomic_fetch_add(p, v, __ATOMIC_RELAXED, __HIP_MEMORY_SCOPE_AGENT);
}

__device__ __forceinline__ v8f wmma4(v2f a, v2f b, v8f c) {
  return __builtin_amdgcn_wmma_f32_16x16x4_f32(false, a, false, b, (short)0, c,
                                               false, false);
}

// ---------- top-k pooling ----------
__global__ void k_scores(const float* __restrict__ x, int F,
                         const float* __restrict__ scorer,
                         float* __restrict__ scores, int n) {
  int i = blockIdx.x * blockDim.x + threadIdx.x;
  if (i >= n) return;
  const float* row = x + (size_t)i * F;
  float dot = 0.f, ss = 0.f;
  for (int k = 0; k < F; ++k) { float w = scorer[k]; dot += row[k] * w; ss += w * w; }
  scores[i] = dot * rsqrtf(ss);
}

// single block, 76 rounds of global argmax (ties -> lowest index, like jax top_k)
__global__ void k_topk(float* __restrict__ scores, int n,
                       float* __restrict__ vals, int* __restrict__ idxs) {
  __shared__ float sv[1024];
  __shared__ int si[1024];
  int tid = threadIdx.x;
  for (int it = 0; it < NH; ++it) {
    float best = -3.402823466e38f; int bi = n;
    for (int j = tid; j < n; j += 1024) {
      float v = scores[j];
      if (v > best) { best = v; bi = j; }   // strided ascending -> first max kept
    }
    sv[tid] = best; si[tid] = bi;
    __syncthreads();
    for (int s = 512; s > 0; s >>= 1) {
      if (tid < s) {
        float v2 = sv[tid + s]; int i2 = si[tid + s];
        if (v2 > sv[tid] || (v2 == sv[tid] && i2 < si[tid])) { sv[tid] = v2; si[tid] = i2; }
      }
      __syncthreads();
    }
    if (tid == 0) { vals[it] = sv[0]; idxs[it] = si[0]; scores[si[0]] = -3.402823466e38f; }
    __syncthreads();
  }
}

// x_tilde[f][j] = x[idx[j]][f] * tanh(vals[j])   -> (F x 76) row-major
__global__ void k_build_xt(const float* __restrict__ x, int F,
                           const int* __restrict__ idxs,
                           const float* __restrict__ vals,
                           float* __restrict__ xt) {
  int i = blockIdx.x * blockDim.x + threadIdx.x;
  if (i >= F * NH) return;
  int f = i / NH, j = i - f * NH;
  xt[i] = x[(size_t)idxs[j] * F + f] * tanhf(vals[j]);
}

// ---------- GRU step on the (F x 76) weight matrix ----------
__global__ void k_gru(const float* __restrict__ xt, const float* __restrict__ Wc,
                      const float* __restrict__ wih, const float* __restrict__ whh,
                      const float* __restrict__ bih, const float* __restrict__ bhh,
                      float* __restrict__ Wn, int F) {
  int i = blockIdx.x * blockDim.x + threadIdx.x;
  if (i >= F * NH) return;
  int row = i / NH, j = i - row * NH;
  const float* xr = xt + (size_t)row * NH;
  const float* hr = Wc + (size_t)row * NH;
  float gx[3], gh[3];
  for (int g = 0; g < 3; ++g) {
    const float* wi = wih + (size_t)(g * NH + j) * NH;
    const float* wh = whh + (size_t)(g * NH + j) * NH;
    float sx = bih[g * NH + j], sh = bhh[g * NH + j];
    for (int k = 0; k < NH; ++k) { sx += xr[k] * wi[k]; sh += hr[k] * wh[k]; }
    gx[g] = sx; gh[g] = sh;
  }
  float r = 1.f / (1.f + expf(-(gx[0] + gh[0])));
  float z = 1.f / (1.f + expf(-(gx[1] + gh[1])));
  float nv = tanhf(gx[2] + r * gh[2]);
  Wn[i] = (1.f - z) * nv + z * hr[j];
}

// ---------- graph conv pieces ----------
__global__ void k_deg(const int* __restrict__ src, const int* __restrict__ dst,
                      float* __restrict__ degO, float* __restrict__ degI, int e) {
  int i = blockIdx.x * blockDim.x + threadIdx.x;
  if (i >= e) return;
  fatomic_add(&degO[src[i]], 1.f);
  fatomic_add(&degI[dst[i]], 1.f);
}

__global__ void k_norm(float* __restrict__ deg, int n) {
  int i = blockIdx.x * blockDim.x + threadIdx.x;
  if (i >= n) return;
  deg[i] = rsqrtf(fmaxf(deg[i], 1.f));   // clip(deg,1)^-0.5
}

// C[M x 76] = rowscale[:,None] * (A[M x K] @ B[K x 76])
// One wave per 16x16 C tile; M must be a multiple of 16 (100000 is).
// B tile staged zero-padded in LDS, pair-interleaved so each lane's B frag is
// one ds_load_b64. A frags are unconditional global v2f loads (rows 8B-aligned).
// K-loop fully unrolled (templated K) -> branch-free, batched loads.
template <int K>
__global__ void k_wmma_gemm(const float* __restrict__ A,
                            const float* __restrict__ B,
                            const float* __restrict__ rowscale,
                            float* __restrict__ C) {
  constexpr int Kp = (K + 3) & ~3;
  constexpr int Kfloor = K & ~3;
  __shared__ float Bs[(Kp / 2) * 32];   // [pair][lane(16)][2]
  const int row0 = blockIdx.x * 16;
  const int col0 = blockIdx.y * 16;
  const int lane = threadIdx.x;
  const int hi = lane >> 4, l = lane & 15;

  for (int idx = lane; idx < Kp * 16; idx += 32) {
    int k = idx >> 4, c = idx & 15;
    int gc = col0 + c;
    float v = (k < K && gc < NH) ? B[k * NH + gc] : 0.f;
    Bs[(k >> 1) * 32 + c * 2 + (k & 1)] = v;
  }
  __syncthreads();

  const int arow = row0 + l;
  const v2f* ap = (const v2f*)(A + (size_t)arow * K);
  v8f acc = {};
#pragma unroll
  for (int k0 = 0; k0 < Kfloor; k0 += 4) {
    v2f a = ap[(k0 >> 1) + hi];
    v2f b = *(const v2f*)&Bs[((k0 >> 1) + hi) * 32 + l * 2];
    acc = wmma4(a, b, acc);
  }
  if constexpr (Kfloor < K) {           // K=166: one guarded epilogue step
    const float* arp = A + (size_t)arow * K;
    int ka = Kfloor + (hi << 1);
    v2f a, b;
    a.x = (ka < K) ? arp[ka] : 0.f;
    a.y = (ka + 1 < K) ? arp[ka + 1] : 0.f;
    b = *(const v2f*)&Bs[((Kfloor >> 1) + hi) * 32 + l * 2];
    acc = wmma4(a, b, acc);
  }
  const int ccol = col0 + l;
  if (ccol < NH) {
    const int rbase = row0 + (hi << 3);
#pragma unroll
    for (int j = 0; j < 8; ++j) {
      int r = rbase + j;
      C[(size_t)r * NH + ccol] = acc[j] * rowscale[r];
    }
  }
}

// agg[dst[e]][f] += h[src[e]][f]   (hardware fp32 atomics, L2-resident agg)
__global__ void k_scatter(const float* __restrict__ h, const int* __restrict__ src,
                          const int* __restrict__ dst, float* __restrict__ agg,
                          int total) {
  int i = blockIdx.x * blockDim.x + threadIdx.x;
  if (i >= total) return;
  int e = i / NH, f = i - e * NH;
  fatomic_add(&agg[(size_t)dst[e] * NH + f], h[(size_t)src[e] * NH + f]);
}

__global__ void k_finalize(const float* __restrict__ agg,
                           const float* __restrict__ normI,
                           float* __restrict__ out, int total) {
  int i = blockIdx.x * blockDim.x + threadIdx.x;
  if (i >= total) return;
  int n = i / NH;
  float v = agg[i] * normI[n];
  out[i] = v >= 0.f ? v : v * RRELU_SLOPE;
}

// ---------- fused MLP head: relu(X@W1+b1)@W2+b2, hidden kept in LDS ----------
// A fragments (19 pairs) loaded once and reused across all 32 column tiles.
// Column tiles 0..30 are branch-free; only the tail tile (cols 496..511) guards.
__global__ void k_mlp(const float* __restrict__ X, const float* __restrict__ W1,
                      const float* __restrict__ b1, const float* __restrict__ W2,
                      const float* __restrict__ b2, float* __restrict__ out, int M) {
  __shared__ float h1[16 * 516];        // pitch 516 -> conflict-free column reads
  const int row0 = blockIdx.x * 16;
  const int lane = threadIdx.x;
  const int hi = lane >> 4, l = lane & 15;
  const int arow = row0 + l;

  const v2f* ap = (const v2f*)(X + (size_t)arow * NH);
  v2f areg[19];
#pragma unroll
  for (int kk = 0; kk < 19; ++kk) areg[kk] = ap[kk * 2 + hi];

  for (int ct = 0; ct < 31; ++ct) {     // cols 0..495, no bounds checks
    const int bcol = ct * 16 + l;
    v8f acc = {};
#pragma unroll
    for (int kk = 0; kk < 19; ++kk) {
      int ka = kk * 4 + (hi << 1);
      v2f b;
      b.x = W1[(size_t)ka * CLSH + bcol];
      b.y = W1[(size_t)(ka + 1) * CLSH + bcol];
      acc = wmma4(areg[kk], b, acc);
    }
    float bias = b1[bcol];
#pragma unroll
    for (int j = 0; j < 8; ++j) {
      float v = acc[j] + bias;
      h1[((hi << 3) + j) * 516 + bcol] = v > 0.f ? v : 0.f;
    }
  }
  {                                     // tail tile: cols 496..511 (510,511 OOB)
    const int bcol = 496 + l;
    const bool cok = bcol < CLSH;
    v8f acc = {};
#pragma unroll
    for (int kk = 0; kk < 19; ++kk) {
      int ka = kk * 4 + (hi << 1);
      v2f b;
      b.x = cok ? W1[(size_t)ka * CLSH + bcol] : 0.f;
      b.y = cok ? W1[(size_t)(ka + 1) * CLSH + bcol] : 0.f;
      acc = wmma4(areg[kk], b, acc);
    }
    float bias = cok ? b1[bcol] : 0.f;
#pragma unroll
    for (int j = 0; j < 8; ++j) {
      float v = acc[j] + bias;
      h1[((hi << 3) + j) * 516 + bcol] = (cok && v > 0.f) ? v : 0.f;
    }
  }
  __syncthreads();
  const int rl = lane >> 1, cls = lane & 1;  // 16 rows x 2 classes = 32 lanes
  float sum = b2[cls];
  const float* hr = h1 + rl * 516;
  for (int k = 0; k < CLSH; ++k) sum += hr[k] * W2[k * 2 + cls];
  out[(size_t)(row0 + rl) * 2 + cls] = sum;
}

extern "C" void kernel_launch(void* const* d_in, const int* in_sizes, int n_in,
                              void* d_out, int out_size, void* d_ws, size_t ws_size,
                              hipStream_t stream) {
  const float* feats = (const float*)d_in[0];
  const int*   src   = (const int*)d_in[1];
  const int*   dst   = (const int*)d_in[2];
  const float* scorer[2]  = {(const float*)d_in[3],  (const float*)d_in[4]};
  const float* gcn_w[2]   = {(const float*)d_in[5],  (const float*)d_in[6]};
  const float* gru_wih[2] = {(const float*)d_in[7],  (const float*)d_in[11]};
  const float* gru_whh[2] = {(const float*)d_in[8],  (const float*)d_in[12]};
  const float* gru_bih[2] = {(const float*)d_in[9],  (const float*)d_in[13]};
  const float* gru_bhh[2] = {(const float*)d_in[10], (const float*)d_in[14]};
  const float* w1 = (const float*)d_in[15];
  const float* b1 = (const float*)d_in[16];
  const float* w2 = (const float*)d_in[17];
  const float* b2 = (const float*)d_in[18];

  // workspace carve-up (~153 MB); every region size is a multiple of 8 bytes
  float* p   = (float*)d_ws;
  float* W_a = p;  p += IN_FEATS * NH;
  float* W_b = p;  p += IN_FEATS * NH;
  float* xt  = p;  p += IN_FEATS * NH;
  float* tvals = p; p += NH;
  int*   tidx  = (int*)p; p += NH;
  float* scores = p; p += NN;
  float* degO = p; p += NN;          // degO/degI contiguous -> one memset
  float* degI = p; p += NN;
  float* h    = p; p += (size_t)NN * NH;
  float* agg  = p; p += (size_t)NN * NH;
  float* fbuf = p; p += (size_t)3 * NN * NH;   // post-layer-0 features per t

  for (int layer = 0; layer < 2; ++layer) {
    const int F = (layer == 0) ? IN_FEATS : NH;
    hipMemcpyAsync(W_a, gcn_w[layer], (size_t)F * NH * sizeof(float),
                   hipMemcpyDeviceToDevice, stream);
    float* Wc = W_a; float* Wn = W_b;
    for (int t = 0; t < TT; ++t) {
      const float* x = (layer == 0) ? feats + (size_t)t * NN * IN_FEATS
                                    : fbuf + (size_t)t * NN * NH;
      // --- top-k pool + GRU weight evolution ---
      k_scores<<<(NN + 255) / 256, 256, 0, stream>>>(x, F, scorer[layer], scores, NN);
      k_topk<<<1, 1024, 0, stream>>>(scores, NN, tvals, tidx);
      k_build_xt<<<(F * NH + 255) / 256, 256, 0, stream>>>(x, F, tidx, tvals, xt);
      k_gru<<<(F * NH + 255) / 256, 256, 0, stream>>>(
          xt, Wc, gru_wih[layer], gru_whh[layer], gru_bih[layer], gru_bhh[layer],
          Wn, F);
      { float* tmp = Wc; Wc = Wn; Wn = tmp; }

      // --- graph conv; in the last layer only t==2 is ever consumed ---
      if (layer == 0 || t == TT - 1) {
        const int* st = src + (size_t)t * EE;
        const int* dt = dst + (size_t)t * EE;
        hipMemsetAsync(degO, 0, (size_t)2 * NN * sizeof(float), stream);
        k_deg<<<(EE + 255) / 256, 256, 0, stream>>>(st, dt, degO, degI, EE);
        k_norm<<<(2 * NN + 255) / 256, 256, 0, stream>>>(degO, 2 * NN);
        dim3 gg(NN / 16, (NH + 15) / 16);
        if (layer == 0)
          k_wmma_gemm<IN_FEATS><<<gg, 32, 0, stream>>>(x, Wc, degO, h);
        else
          k_wmma_gemm<NH><<<gg, 32, 0, stream>>>(x, Wc, degO, h);
        hipMemsetAsync(agg, 0, (size_t)NN * NH * sizeof(float), stream);
        const int tot = EE * NH;   // 121.6M < 2^31
        k_scatter<<<(tot + 255) / 256, 256, 0, stream>>>(h, st, dt, agg, tot);
        // safe to overwrite fbuf[t]: its only readers (scores/xt/gemm) already ran
        float* outp = fbuf + (size_t)t * NN * NH;
        k_finalize<<<(NN * NH + 255) / 256, 256, 0, stream>>>(agg, degI, outp, NN * NH);
      }
    }
  }
  // MLP head on final-timestep features
  k_mlp<<<NN / 16, 32, 0, stream>>>(fbuf + (size_t)2 * NN * NH, w1, b1, w2, b2,
                                    (float*)d_out, NN);
}